// PendulumSolver_687194767674
// MI455X (gfx1250) — compile-verified
//
#include <hip/hip_runtime.h>
#include <stdint.h>

// Pendulum RK integrator for MI455X (gfx1250, wave32).
// One thread per pendulum; LDS-staged transpose; async LDS->global writeback.

#define LEN   2048          // episode length (timesteps)
#define TS    32            // timesteps per LDS tile
#define TPB   64            // threads per block (2 wave32)
#define PAD   33            // LDS row pitch (floats), coprime with 64 banks
#define TWO_PI 6.28318530717958647692f

__global__ __launch_bounds__(TPB, 1)
void pendulum_rk_kernel(const float* __restrict__ init_conds,   // (B,2)
                        const float* __restrict__ params,       // (B,4)
                        float* __restrict__ out)                // (B,LEN)
{
    __shared__ float tile[2][TPB * PAD];   // double-buffered: 2 * 64 * 33 * 4 = 16.9 KB

    const int tid  = threadIdx.x;
    const int lane = tid & 31;
    const int wid  = tid >> 5;
    const int b    = blockIdx.x * TPB + tid;

    // --- per-pendulum state & constants (registers) ---
    float th = init_conds[2 * b + 0];
    float w  = init_conds[2 * b + 1];
    const float om   = params[4 * b + 0];
    const float ga   = params[4 * b + 1];
    const float A    = params[4 * b + 2];
    const float ph   = params[4 * b + 3];
    const float om2  = om * om;
    const float Aom2 = A * om2;
    const float c    = TWO_PI * ph;
    const float dt   = 0.01f;
    const float dt3  = dt * (1.0f / 3.0f);

    // --- wave-local flush mapping: 8 lanes cover one pendulum row (8 x float4 = 32 floats)
    const int p0 = lane >> 3;          // row sub-index, advances by 4 each flush iter
    const int q  = lane & 7;           // float4 slot within the 32-float row segment
    const int prow = wid * 32;         // first LDS row owned by this wave

    // LDS byte offset of tile base (low 32 bits of the generic pointer == LDS offset)
    const uint32_t lds0 = (uint32_t)(uintptr_t)(&tile[0][0]);

    // Global base for this lane's stores: row (block_base + prow + p0), column q*4
    float* const out_base =
        out + ((size_t)(blockIdx.x * TPB + prow + p0)) * LEN + (size_t)(q * 4);

    const int NT = LEN / TS;   // 64 tiles
    for (int k = 0; k < NT; ++k) {
        const int cur = k & 1;
        float* buf = &tile[cur][0];

        // Tile k-2 used this buffer; allow tile k-1's 8 async stores to remain in flight.
        asm volatile("s_wait_asynccnt 8" ::: "memory");

        #pragma unroll 1
        for (int s = 0; s < TS; ++s) {
            // record sample j = k*TS + s (theta at time t_j)
            buf[tid * PAD + s] = th;

            const float ti = (float)(k * TS + s) * dt;

            // driving force at the 4 stage times (independent of state -> early ILP)
            const float f1 = Aom2 * __cosf(c * ti);
            const float f2 = Aom2 * __cosf(c * (ti + dt3));
            const float f3 = Aom2 * __cosf(c * (ti + 2.0f * dt3));
            const float f4 = Aom2 * __cosf(c * (ti + dt));

            // RK stages (matches reference step() exactly)
            const float k1t = w;
            const float k1w = f1 - ga * w - om2 * __sinf(th);

            const float th2 = th + dt3 * k1t;
            const float w2  = w  + dt3 * k1w;
            const float k2t = w2;
            const float k2w = f2 - ga * w2 - om2 * __sinf(th2);

            const float th3 = th + dt * (k2t - (1.0f / 3.0f) * k1t);
            const float w3  = w  + dt * (k2w - (1.0f / 3.0f) * k1w);
            const float k3t = w3;
            const float k3w = f3 - ga * w3 - om2 * __sinf(th3);

            const float th4 = th + dt * (k1t - k2t + k3t);
            const float w4  = w  + dt * (k1w - k2w + k3w);
            const float k4t = w4;
            const float k4w = f4 - ga * w4 - om2 * __sinf(th4);

            th += dt * 0.125f * (k1t + 3.0f * (k2t + k3t) + k4t);
            w  += dt * 0.125f * (k1w + 3.0f * (k2w + k3w) + k4w);
            // (the step after the last recorded sample is computed but never stored)
        }

        // Ensure this wave's LDS stores have landed before the async engine reads them.
        asm volatile("s_wait_dscnt 0" ::: "memory");

        // Async flush: 8 instructions x (32 lanes x 16B) = 4 KB tile, fully coalesced
        // (8 consecutive lanes -> one contiguous 128B row segment in `out`).
        uint32_t lbase = lds0 + (uint32_t)cur * (uint32_t)(TPB * PAD * 4)
                       + (uint32_t)(((prow + p0) * PAD + q * 4) * 4);
        float* gaddr = out_base + (size_t)k * TS;
        #pragma unroll
        for (int i = 0; i < 8; ++i) {
            asm volatile("global_store_async_from_lds_b128 %0, %1, off"
                         :
                         : "v"(gaddr), "v"(lbase)
                         : "memory");
            gaddr += 4 * LEN;          // +4 output rows
            lbase += 4 * PAD * 4;      // +4 LDS rows (bytes)
        }
    }

    // Drain all outstanding async stores before wave exit.
    asm volatile("s_wait_asynccnt 0" ::: "memory");
}

extern "C" void kernel_launch(void* const* d_in, const int* in_sizes, int n_in,
                              void* d_out, int out_size, void* d_ws, size_t ws_size,
                              hipStream_t stream) {
    const float* init_conds = (const float*)d_in[0];   // (B,2) f32
    const float* params     = (const float*)d_in[1];   // (B,4) f32
    float* out              = (float*)d_out;           // (B,LEN) f32

    const int B = in_sizes[0] / 2;                     // 16384
    dim3 grid(B / TPB), block(TPB);
    pendulum_rk_kernel<<<grid, block, 0, stream>>>(init_conds, params, out);
}